// VGAE_22204980921072
// MI455X (gfx1250) — compile-verified
//
#include <hip/hip_runtime.h>

typedef float v2f __attribute__((ext_vector_type(2)));
typedef float v8f __attribute__((ext_vector_type(8)));

#define N_NODES 10000
#define F_IN    512
#define H1      256
#define H2      64
#define N_EDGES 320000
#define MTILES  625          // N_NODES / 16

#define WMMA_F32(A, B, C) \
    __builtin_amdgcn_wmma_f32_16x16x4_f32(false, (A), false, (B), (short)0, (C), false, false)

// ---------------------------------------------------------------------------
// GEMM1: H0[10000x256] = x[10000x512] @ W1[512x256]   (f32 WMMA 16x16x4)
// 2x2 tiles per wave (32x32 region). grid.x = ceil(625/2)=313 (M-pairs,
// clamped), threadIdx.y = N-group (8 groups of 32 cols). block = (32,8).
// ---------------------------------------------------------------------------
__global__ __launch_bounds__(256) void k_gemm_xw1(const float* __restrict__ X,
                                                  const float* __restrict__ W1,
                                                  float* __restrict__ H0) {
    const int lane = threadIdx.x & 31;
    const int half = lane >> 4;
    const int lr   = lane & 15;

    const int mt0 = blockIdx.x * 2;                    // 0..624 (even)
    const int mt1 = mt0 + 1;
    const int m1c = (mt1 > MTILES - 1) ? MTILES - 1 : mt1;  // scalar clamp
    const int n0  = threadIdx.y * 32;                  // col base, 0..224

    const float* arow0 = X + (long long)(mt0 * 16 + lr) * F_IN + 2 * half;
    const float* arow1 = X + (long long)(m1c * 16 + lr) * F_IN + 2 * half;
    const float* bcol0 = W1 + (long long)(2 * half) * H1 + n0 + lr;
    const float* bcol1 = bcol0 + 16;

    v8f c00 = {}, c01 = {}, c10 = {}, c11 = {};
#pragma unroll 4
    for (int k = 0; k < F_IN; k += 4) {
        v2f a0; a0.x = arow0[k]; a0.y = arow0[k + 1];
        v2f a1; a1.x = arow1[k]; a1.y = arow1[k + 1];
        v2f b0; b0.x = bcol0[(long long)k * H1]; b0.y = bcol0[(long long)(k + 1) * H1];
        v2f b1; b1.x = bcol1[(long long)k * H1]; b1.y = bcol1[(long long)(k + 1) * H1];
        c00 = WMMA_F32(a0, b0, c00);
        c01 = WMMA_F32(a0, b1, c01);
        c10 = WMMA_F32(a1, b0, c10);
        c11 = WMMA_F32(a1, b1, c11);
    }
#pragma unroll
    for (int v = 0; v < 8; ++v) {
        long long r0 = (long long)(mt0 * 16 + v + 8 * half) * H1;
        H0[r0 + n0 + lr]      = c00[v];
        H0[r0 + n0 + 16 + lr] = c01[v];
    }
    if (mt1 < MTILES) {
#pragma unroll
        for (int v = 0; v < 8; ++v) {
            long long r1 = (long long)(mt1 * 16 + v + 8 * half) * H1;
            H0[r1 + n0 + lr]      = c10[v];
            H0[r1 + n0 + 16 + lr] = c11[v];
        }
    }
}

// ---------------------------------------------------------------------------
// SpMM 1: hidden[dst] += w * H0[src]  over 320k edges, 256 cols.
// ---------------------------------------------------------------------------
__global__ void k_spmm1(const int* __restrict__ ei, const float* __restrict__ ew,
                        const float* __restrict__ H0, float* __restrict__ Hid) {
    unsigned t = blockIdx.x * blockDim.x + threadIdx.x;
    int e = t >> 6;
    int c = (t & 63) * 4;
    if (e >= N_EDGES) return;
    int   src = ei[e];
    int   dst = ei[N_EDGES + e];
    float w   = ew[e];
    const float4 v = *(const float4*)(H0 + (long long)src * H1 + c);
    float* o = Hid + (long long)dst * H1 + c;
    atomicAdd(o + 0, w * v.x);
    atomicAdd(o + 1, w * v.y);
    atomicAdd(o + 2, w * v.z);
    atomicAdd(o + 3, w * v.w);
}

// ---------------------------------------------------------------------------
// GEMM2+3 fused: T2 = relu(Hid)@W2, T3 = relu(Hid)@W3.  K=256, full N=64
// per wave (4 col-tiles x 2 matrices = 8 accumulators, shared relu'd A).
// Hidden is read exactly once. block = (32,8); tasks = 625 M-tiles.
// ---------------------------------------------------------------------------
__global__ __launch_bounds__(256) void k_gemm_h23(const float* __restrict__ Hid,
                                                  const float* __restrict__ W2,
                                                  const float* __restrict__ W3,
                                                  float* __restrict__ T2,
                                                  float* __restrict__ T3) {
    const int task = blockIdx.x * blockDim.y + threadIdx.y;
    if (task >= MTILES) return;                 // whole-wave uniform exit
    const int lane = threadIdx.x & 31;
    const int half = lane >> 4;
    const int lr   = lane & 15;
    const int m0   = task * 16;

    const float* arow = Hid + (long long)(m0 + lr) * H1 + 2 * half;

    v8f c2[4] = {}, c3[4] = {};
#pragma unroll 4
    for (int k = 0; k < H1; k += 4) {
        v2f a;
        a.x = fmaxf(arow[k], 0.0f);
        a.y = fmaxf(arow[k + 1], 0.0f);
        const int r0 = (k + 2 * half) * H2 + lr;
#pragma unroll
        for (int j = 0; j < 4; ++j) {
            v2f b2; b2.x = W2[r0 + j * 16]; b2.y = W2[r0 + H2 + j * 16];
            v2f b3; b3.x = W3[r0 + j * 16]; b3.y = W3[r0 + H2 + j * 16];
            c2[j] = WMMA_F32(a, b2, c2[j]);
            c3[j] = WMMA_F32(a, b3, c3[j]);
        }
    }
#pragma unroll
    for (int v = 0; v < 8; ++v) {
        long long row = (long long)(m0 + v + 8 * half) * H2;
#pragma unroll
        for (int j = 0; j < 4; ++j) {
            T2[row + j * 16 + lr] = c2[j][v];
            T3[row + j * 16 + lr] = c3[j][v];
        }
    }
}

// ---------------------------------------------------------------------------
// SpMM 2 (both heads): zm[dst] += w*T2[src]; zl[dst] += w*T3[src].  64 cols.
// ---------------------------------------------------------------------------
__global__ void k_spmm2(const int* __restrict__ ei, const float* __restrict__ ew,
                        const float* __restrict__ T2, const float* __restrict__ T3,
                        float* __restrict__ ZM, float* __restrict__ ZL) {
    unsigned t = blockIdx.x * blockDim.x + threadIdx.x;
    int e = t >> 4;
    int c = (t & 15) * 4;
    if (e >= N_EDGES) return;
    int   src = ei[e];
    int   dst = ei[N_EDGES + e];
    float w   = ew[e];
    const float4 v2 = *(const float4*)(T2 + (long long)src * H2 + c);
    const float4 v3 = *(const float4*)(T3 + (long long)src * H2 + c);
    float* om = ZM + (long long)dst * H2 + c;
    float* ol = ZL + (long long)dst * H2 + c;
    atomicAdd(om + 0, w * v2.x); atomicAdd(om + 1, w * v2.y);
    atomicAdd(om + 2, w * v2.z); atomicAdd(om + 3, w * v2.w);
    atomicAdd(ol + 0, w * v3.x); atomicAdd(ol + 1, w * v3.y);
    atomicAdd(ol + 2, w * v3.z); atomicAdd(ol + 3, w * v3.w);
}

// ---------------------------------------------------------------------------
// Reparameterization: z = zm + exp(zl) * noise   (640000 elems)
// ---------------------------------------------------------------------------
__global__ void k_reparam(const float* __restrict__ ZM, const float* __restrict__ ZL,
                          const float* __restrict__ NS, float* __restrict__ Z) {
    int i = blockIdx.x * blockDim.x + threadIdx.x;
    if (i < N_NODES * H2)
        Z[i] = ZM[i] + __expf(ZL[i]) * NS[i];
}

// ---------------------------------------------------------------------------
// Decode: Out = sigmoid(Z @ Z^T), 10000x10000, K=64.  Dominant kernel.
// 4x2 WMMA tiles per wave (64x32 output): fragment reads drop from 8B to
// 3B per stored byte (L2-read roofline). Tile indices clamped (scalar,
// wave-uniform => EXEC stays all-ones); stores guarded on unclamped index.
// Output streamed with non-temporal stores (400MB, never re-read; keeps
// z resident in the 192MB L2).
// ---------------------------------------------------------------------------
#define DEC_MG 157    // ceil(625/4)
#define DEC_NG 313    // ceil(625/2)

__global__ __launch_bounds__(256) void k_decode(const float* __restrict__ Z,
                                                float* __restrict__ Out) {
    const long long task = (long long)blockIdx.x * blockDim.y + threadIdx.y;
    if (task >= (long long)DEC_MG * DEC_NG) return;   // whole-wave uniform exit
    const int lane = threadIdx.x & 31;
    const int half = lane >> 4;
    const int lr   = lane & 15;
    const int mg = (int)(task / DEC_NG);
    const int ng = (int)(task % DEC_NG);

    int mt[4], nt[2];
    const float* arow[4];
    const float* brow[2];
#pragma unroll
    for (int i = 0; i < 4; ++i) {
        mt[i] = mg * 4 + i;
        int mc = (mt[i] > MTILES - 1) ? MTILES - 1 : mt[i];
        arow[i] = Z + (long long)(mc * 16 + lr) * H2 + 2 * half;
    }
#pragma unroll
    for (int j = 0; j < 2; ++j) {
        nt[j] = ng * 2 + j;
        int nc = (nt[j] > MTILES - 1) ? MTILES - 1 : nt[j];
        brow[j] = Z + (long long)(nc * 16 + lr) * H2 + 2 * half;
    }

    v8f c[4][2] = {};
#pragma unroll
    for (int k = 0; k < H2; k += 4) {
        v2f a[4], b[2];
#pragma unroll
        for (int i = 0; i < 4; ++i) { a[i].x = arow[i][k]; a[i].y = arow[i][k + 1]; }
#pragma unroll
        for (int j = 0; j < 2; ++j) { b[j].x = brow[j][k]; b[j].y = brow[j][k + 1]; }
#pragma unroll
        for (int i = 0; i < 4; ++i)
#pragma unroll
            for (int j = 0; j < 2; ++j)
                c[i][j] = WMMA_F32(a[i], b[j], c[i][j]);
    }

#pragma unroll
    for (int i = 0; i < 4; ++i) {
        if (mt[i] >= MTILES) continue;
#pragma unroll
        for (int v = 0; v < 8; ++v) {
            long long row = (long long)(mt[i] * 16 + v + 8 * half) * N_NODES;
#pragma unroll
            for (int j = 0; j < 2; ++j) {
                if (nt[j] >= MTILES) continue;
                float s = 1.0f / (1.0f + __expf(-c[i][j][v]));
                __builtin_nontemporal_store(s, Out + row + nt[j] * 16 + lr);
            }
        }
    }
}

// ---------------------------------------------------------------------------
extern "C" void kernel_launch(void* const* d_in, const int* in_sizes, int n_in,
                              void* d_out, int out_size, void* d_ws, size_t ws_size,
                              hipStream_t stream) {
    const float* x     = (const float*)d_in[0];   // 10000 x 512
    const float* W1    = (const float*)d_in[1];   // 512 x 256
    const float* W2    = (const float*)d_in[2];   // 256 x 64
    const float* W3    = (const float*)d_in[3];   // 256 x 64
    const float* ew    = (const float*)d_in[4];   // 320000
    const float* noise = (const float*)d_in[5];   // 10000 x 64
    const int*   ei    = (const int*)d_in[6];     // 2 x 320000 (src row, dst row)

    float* ws     = (float*)d_ws;
    float* hidden = ws;                 // 2,560,000 f  (spmm1 accumulator)
    float* h0     = ws + 2560000;       // 2,560,000 f  (x@W1)
    float* t2     = ws + 5120000;       //   640,000 f
    float* t3     = ws + 5760000;       //   640,000 f
    float* zm     = ws + 6400000;       //   640,000 f  (atomic accum)
    float* zl     = ws + 7040000;       //   640,000 f  (atomic accum)
    float* z      = t2;                 // reuse t2 slot after spmm2

    // zero the atomic accumulators every call (graph-capture-safe)
    hipMemsetAsync(hidden, 0, (size_t)2560000 * sizeof(float), stream);
    hipMemsetAsync(zm,     0, (size_t)640000  * sizeof(float), stream);
    hipMemsetAsync(zl,     0, (size_t)640000  * sizeof(float), stream);

    // 1) h0 = x @ W1
    k_gemm_xw1<<<dim3((MTILES + 1) / 2), dim3(32, 8), 0, stream>>>(x, W1, h0);
    // 2) hidden = spmm(adj, h0)   (relu deferred into next GEMM's loads)
    k_spmm1<<<(N_EDGES * 64) / 256, 256, 0, stream>>>(ei, ew, h0, hidden);
    // 3) t2 = relu(hidden)@W2 ; t3 = relu(hidden)@W3   (fused, full N per wave)
    k_gemm_h23<<<(MTILES + 7) / 8, dim3(32, 8), 0, stream>>>(hidden, W2, W3, t2, t3);
    // 4) zm = spmm(adj, t2) ; zl = spmm(adj, t3)   (fused)
    k_spmm2<<<(N_EDGES * 16) / 256, 256, 0, stream>>>(ei, ew, t2, t3, zm, zl);
    // 5) z = zm + exp(zl) * noise
    k_reparam<<<(N_NODES * H2 + 255) / 256, 256, 0, stream>>>(zm, zl, noise, z);
    // 6) Out = sigmoid(z @ z^T), 4x2 tiles per wave
    const long long tasks = (long long)DEC_MG * DEC_NG;   // 49141
    k_decode<<<(unsigned)((tasks + 7) / 8), dim3(32, 8), 0, stream>>>(z, (float*)d_out);
}